// ReOrthLayer_71528385347745
// MI455X (gfx1250) — compile-verified
//
#include <hip/hip_runtime.h>

// Re-orthonormalization (thin QR, Q factor only) of 8192 x (256x16) f32 matrices.
// CholeskyQR2 per matrix, one wave32 per matrix. X staged once into LDS via
// CDNA5 async global->LDS DMA, all GEMM work on V_WMMA_F32_16X16X4_F32 with
// dual accumulators, Q written once to global. Memory-bound: ~268 MB @ 23.3 TB/s.

#define NBATCH 8192
#define DROWS  256
#define QCOLS  16

typedef __attribute__((ext_vector_type(2))) float v2f;
typedef __attribute__((ext_vector_type(8))) float v8f;

__device__ __forceinline__ v8f wmma_f32_16x16x4(v2f a, v2f b, v8f c) {
  // 8 args: (neg_a, A, neg_b, B, c_mod, C, reuse_a, reuse_b)
  return __builtin_amdgcn_wmma_f32_16x16x4_f32(
      false, a, false, b, (short)0, c, false, false);
}

extern "C" __global__ __launch_bounds__(32)
void reorth_qr_kernel(const float* __restrict__ X, float* __restrict__ Out)
{
  // 16 KB tile of the matrix + two padded 16x16 scratch matrices
  __shared__ float sX[DROWS * QCOLS];   // row-major, stride 16
  __shared__ float sR[QCOLS * 17];      // Gram -> Cholesky R (upper), pad 17
  __shared__ float sRi[QCOLS * 17];     // R^{-1} (upper), pad 17

  const int b    = blockIdx.x;
  const int l    = threadIdx.x;   // 0..31
  const int half = l >> 4;        // 0: lanes 0-15, 1: lanes 16-31
  const int lm   = l & 15;

  const float* __restrict__ Xb = X   + (size_t)b * (DROWS * QCOLS);
  float*       __restrict__ Ob = Out + (size_t)b * (DROWS * QCOLS);

  // ---- Stage X into LDS with async DMA: 16 KB, 16 B/lane/op, 32 ops ----
  // GLOBAL_LOAD_ASYNC_TO_LDS_B128, GV mode: VDST = LDS byte addr, VADDR = 64b
  // global addr. Tracked by ASYNCcnt; wait before first LDS consumer.
  {
    const float* gp = Xb + l * 4;                       // 16 B per lane
    unsigned lds = (unsigned)(uintptr_t)(sX) + (unsigned)(l * 16);
    #pragma unroll 8
    for (int i = 0; i < 32; ++i) {
      asm volatile("global_load_async_to_lds_b128 %0, %1, off"
                   :: "v"(lds), "v"(gp) : "memory");
      gp  += 128;     // 32 lanes * 4 floats
      lds += 512;     // 32 lanes * 16 bytes
    }
    asm volatile("s_wait_asynccnt 0x0" ::: "memory");
  }
  __syncthreads();

  for (int pass = 0; pass < 2; ++pass) {
    // ================= Gram: G = S^T S  (S = sX, 256x16) =================
    // A-frag: lane=M(=q col), {vgpr,half}=K(row).  B-frag: lane=N, same K map.
    // For S^T S both fragments contain S[r0+k][lane&15] -> one load, used twice.
    // Two accumulators to split the WMMA C-chain dependency.
    v8f g0 = {}, g1 = {};
    for (int c = 0; c < DROWS / 8; ++c) {
      const int r0 = c * 8 + half * 2;
      v2f a0, a1;
      a0.x = sX[(r0    ) * QCOLS + lm];
      a0.y = sX[(r0 + 1) * QCOLS + lm];
      a1.x = sX[(r0 + 4) * QCOLS + lm];
      a1.y = sX[(r0 + 5) * QCOLS + lm];
      g0 = wmma_f32_16x16x4(a0, a0, g0);
      g1 = wmma_f32_16x16x4(a1, a1, g1);
    }
    const v8f g = g0 + g1;
    // C/D layout: vgpr v, lanes 0-15 -> row v, lanes 16-31 -> row v+8, col=lane&15
    #pragma unroll
    for (int v = 0; v < 8; ++v)
      sR[(v + 8 * half) * 17 + lm] = g[v];
    __syncthreads();

    // ================= Cholesky (upper R) of G, in sR =================
    for (int k = 0; k < QCOLS; ++k) {
      if (l == k) sR[k * 17 + k] = sqrtf(sR[k * 17 + k]);
      __syncthreads();
      const float rkk = sR[k * 17 + k];
      if (l > k && l < QCOLS) sR[k * 17 + l] = sR[k * 17 + l] / rkk;
      __syncthreads();
      if (l > k && l < QCOLS) {
        const float rkl = sR[k * 17 + l];
        for (int i = k + 1; i <= l; ++i)
          sR[i * 17 + l] -= sR[k * 17 + i] * rkl;
      }
      __syncthreads();
    }

    // ================= Rinv = R^{-1} (upper), lane j solves column j =====
    if (l < QCOLS) {
      const int j = l;
      for (int i = j + 1; i < QCOLS; ++i) sRi[i * 17 + j] = 0.0f;
      sRi[j * 17 + j] = 1.0f / sR[j * 17 + j];
      for (int i = j - 1; i >= 0; --i) {
        float s = 0.0f;
        for (int t = i + 1; t <= j; ++t)
          s += sR[i * 17 + t] * sRi[t * 17 + j];
        sRi[i * 17 + j] = -s / sR[i * 17 + i];
      }
    }
    __syncthreads();

    // ================= S = S * Rinv  (16 row-blocks of 16x16) =============
    for (int rb = 0; rb < DROWS / 16; ++rb) {
      v8f acc0 = {}, acc1 = {};
      #pragma unroll
      for (int kp = 0; kp < 2; ++kp) {          // kc = kp and kp+2, independent
        const int ka = kp * 4 + half * 2;       // kc = kp
        const int kb = (kp + 2) * 4 + half * 2; // kc = kp+2
        v2f aa, ab, ba, bb;
        aa.x = sX[(rb * 16 + lm) * QCOLS + ka];
        aa.y = sX[(rb * 16 + lm) * QCOLS + ka + 1];
        ab.x = sX[(rb * 16 + lm) * QCOLS + kb];
        ab.y = sX[(rb * 16 + lm) * QCOLS + kb + 1];
        ba.x = sRi[(ka    ) * 17 + lm];
        ba.y = sRi[(ka + 1) * 17 + lm];
        bb.x = sRi[(kb    ) * 17 + lm];
        bb.y = sRi[(kb + 1) * 17 + lm];
        acc0 = wmma_f32_16x16x4(aa, ba, acc0);
        acc1 = wmma_f32_16x16x4(ab, bb, acc1);
      }
      const v8f acc = acc0 + acc1;
      if (pass == 0) {
        #pragma unroll
        for (int v = 0; v < 8; ++v)
          sX[(rb * 16 + v + 8 * half) * QCOLS + lm] = acc[v];
      } else {
        #pragma unroll
        for (int v = 0; v < 8; ++v)
          Ob[(size_t)(rb * 16 + v + 8 * half) * QCOLS + lm] = acc[v];
      }
    }
    __syncthreads();
  }
}

extern "C" void kernel_launch(void* const* d_in, const int* in_sizes, int n_in,
                              void* d_out, int out_size, void* d_ws, size_t ws_size,
                              hipStream_t stream) {
  (void)in_sizes; (void)n_in; (void)out_size; (void)d_ws; (void)ws_size;
  const float* X = (const float*)d_in[0];
  float* Outp = (float*)d_out;
  reorth_qr_kernel<<<dim3(NBATCH), dim3(32), 0, stream>>>(X, Outp);
}